// MultiScaleAttn_54030688584235
// MI455X (gfx1250) — compile-verified
//
#include <hip/hip_runtime.h>
#include <hip/hip_bf16.h>
#include <math.h>

// ---------------------------------------------------------------------------
// CDNA5 (gfx1250) implementation. All big GEMM-like ops (1x1 convs, 3x3 convs,
// text-linear, emb x t^T score GEMM, IPA projections) run through
// V_WMMA_F32_16X16X4_F32 (full fp32 precision). wave32 layouts per ISA 7.12.2.
// Round 2: 32-bit marching address offsets, uniform full-tile fast path,
// software-pipelined inner loop, global_prefetch_b8 hints.
// ---------------------------------------------------------------------------

typedef float v2f __attribute__((ext_vector_type(2)));
typedef float v8f __attribute__((ext_vector_type(8)));

#define BN_RS 0.9995003746877732f   /* 1/sqrt(1+0.001) */
#define LN_EPS 1e-5f

__device__ __forceinline__ float silu_f(float x) { return x / (1.0f + __expf(-x)); }

// ===========================================================================
// Generic strided 1x1-conv / GEMM:
//   Y[b,co,n] = act(s[co]*BN_RS * sum_ci W[b?][co,ci]*X[b,ci,n] + bias[co])
// X element (b,ci,n) at X + b*xbs + ci*xCiStr + n*xNStr   (all int strides)
// Y element (b,co,n) at Y + b*ybs + co*yRowStr + n*yColStr
// One wave computes a 16(M=co) x 64(N) tile via 4 accumulators.
// Requires: Co % 16 == 0, Cin % 4 == 0. Block = 128 threads = 4 waves.
// ===========================================================================
__global__ __launch_bounds__(128) void conv1x1_wmma(
    const float* __restrict__ X, int xbs, int xCiStr, int xNStr,
    const float* __restrict__ Wt, int wbs,
    const float* __restrict__ scale, const float* __restrict__ bias,
    float* __restrict__ Y, int ybs, int yRowStr, int yColStr,
    int Cin, int Co, int N, int act)
{
    const int wave = threadIdx.x >> 5;
    const int lane = threadIdx.x & 31;
    const int l15  = lane & 15;
    const int hi   = lane >> 4;          // 0 or 1 (half-wave)
    const int coBase = blockIdx.y * 64 + wave * 16;
    if (coBase >= Co) return;            // wave-uniform exit, EXEC stays full
    const int b = blockIdx.z;
    const int nBase = blockIdx.x * 64;

    const float* __restrict__ Xb = X + (size_t)b * xbs;   // uniform base
    const float* __restrict__ Wb = Wt + (size_t)b * wbs;  // uniform base
    const int kbase = hi * 2;            // A/B K-layout: lanes 0-15 K0/K1, 16-31 K2/K3

    v8f acc0 = {0,0,0,0,0,0,0,0};
    v8f acc1 = {0,0,0,0,0,0,0,0};
    v8f acc2 = {0,0,0,0,0,0,0,0};
    v8f acc3 = {0,0,0,0,0,0,0,0};

    int aoff = (coBase + l15) * Cin + kbase;     // marches +4
    const int bstep = 4 * xCiStr;

    if (nBase + 64 <= N) {
        // ---- full tile: no masks, 2-stage software pipeline ----
        int b0o = (nBase +  0 + l15) * xNStr + kbase * xCiStr;
        int b1o = (nBase + 16 + l15) * xNStr + kbase * xCiStr;
        int b2o = (nBase + 32 + l15) * xNStr + kbase * xCiStr;
        int b3o = (nBase + 48 + l15) * xNStr + kbase * xCiStr;

        v2f a, f0, f1, f2, f3;
        a.x  = Wb[aoff];          a.y  = Wb[aoff + 1];
        f0.x = Xb[b0o];           f0.y = Xb[b0o + xCiStr];
        f1.x = Xb[b1o];           f1.y = Xb[b1o + xCiStr];
        f2.x = Xb[b2o];           f2.y = Xb[b2o + xCiStr];
        f3.x = Xb[b3o];           f3.y = Xb[b3o + xCiStr];

        for (int k0 = 4; k0 < Cin; k0 += 4) {
            aoff += 4; b0o += bstep; b1o += bstep; b2o += bstep; b3o += bstep;
            // prefetch the streams two K-steps ahead (speculative, drop on fault)
            __builtin_prefetch(&Xb[b0o + 2 * bstep], 0, 1);
            __builtin_prefetch(&Xb[b2o + 2 * bstep], 0, 1);
            v2f an, g0, g1, g2, g3;
            an.x = Wb[aoff];      an.y = Wb[aoff + 1];
            g0.x = Xb[b0o];       g0.y = Xb[b0o + xCiStr];
            g1.x = Xb[b1o];       g1.y = Xb[b1o + xCiStr];
            g2.x = Xb[b2o];       g2.y = Xb[b2o + xCiStr];
            g3.x = Xb[b3o];       g3.y = Xb[b3o + xCiStr];
            acc0 = __builtin_amdgcn_wmma_f32_16x16x4_f32(false, a, false, f0, (short)0, acc0, false, false);
            acc1 = __builtin_amdgcn_wmma_f32_16x16x4_f32(false, a, false, f1, (short)0, acc1, false, false);
            acc2 = __builtin_amdgcn_wmma_f32_16x16x4_f32(false, a, false, f2, (short)0, acc2, false, false);
            acc3 = __builtin_amdgcn_wmma_f32_16x16x4_f32(false, a, false, f3, (short)0, acc3, false, false);
            a = an; f0 = g0; f1 = g1; f2 = g2; f3 = g3;
        }
        acc0 = __builtin_amdgcn_wmma_f32_16x16x4_f32(false, a, false, f0, (short)0, acc0, false, false);
        acc1 = __builtin_amdgcn_wmma_f32_16x16x4_f32(false, a, false, f1, (short)0, acc1, false, false);
        acc2 = __builtin_amdgcn_wmma_f32_16x16x4_f32(false, a, false, f2, (short)0, acc2, false, false);
        acc3 = __builtin_amdgcn_wmma_f32_16x16x4_f32(false, a, false, f3, (short)0, acc3, false, false);
    } else {
        // ---- edge tile: clamped addresses + value masks ----
        float msk[4];
        int   boff[4];
#pragma unroll
        for (int j = 0; j < 4; ++j) {
            int c = nBase + j * 16 + l15;
            msk[j] = (c < N) ? 1.0f : 0.0f;
            if (c >= N) c = N - 1;
            boff[j] = c * xNStr + kbase * xCiStr;
        }
        for (int k0 = 0; k0 < Cin; k0 += 4) {
            v2f a, f0, f1, f2, f3;
            a.x  = Wb[aoff];              a.y  = Wb[aoff + 1];
            f0.x = Xb[boff[0]] * msk[0];  f0.y = Xb[boff[0] + xCiStr] * msk[0];
            f1.x = Xb[boff[1]] * msk[1];  f1.y = Xb[boff[1] + xCiStr] * msk[1];
            f2.x = Xb[boff[2]] * msk[2];  f2.y = Xb[boff[2] + xCiStr] * msk[2];
            f3.x = Xb[boff[3]] * msk[3];  f3.y = Xb[boff[3] + xCiStr] * msk[3];
            acc0 = __builtin_amdgcn_wmma_f32_16x16x4_f32(false, a, false, f0, (short)0, acc0, false, false);
            acc1 = __builtin_amdgcn_wmma_f32_16x16x4_f32(false, a, false, f1, (short)0, acc1, false, false);
            acc2 = __builtin_amdgcn_wmma_f32_16x16x4_f32(false, a, false, f2, (short)0, acc2, false, false);
            acc3 = __builtin_amdgcn_wmma_f32_16x16x4_f32(false, a, false, f3, (short)0, acc3, false, false);
            aoff += 4;
            boff[0] += bstep; boff[1] += bstep; boff[2] += bstep; boff[3] += bstep;
        }
    }

    float* __restrict__ Yb = Y + (size_t)b * ybs;
    v8f accs[4] = {acc0, acc1, acc2, acc3};
#pragma unroll
    for (int r = 0; r < 8; ++r) {
        const int row = coBase + hi * 8 + r;   // D layout: VGPR r -> M=r / M=8+r
        const float sc = scale ? scale[row] * BN_RS : 1.0f;
        const float bb = bias ? bias[row] : 0.0f;
#pragma unroll
        for (int j = 0; j < 4; ++j) {
            const int col = nBase + j * 16 + l15;
            if (col < N) {
                float v = accs[j][r] * sc + bb;
                if (act) v = silu_f(v);
                Yb[row * yRowStr + col * yColStr] = v;
            }
        }
    }
}

// ===========================================================================
// 3x3 conv (pad=1), NCHW. One wave: 16 co x 16 x-pixels of one image row.
// y = act(s*conv + bias) [* awmul[b,pix]] [+ resid[b,co,pix]]
// ===========================================================================
__global__ __launch_bounds__(128) void conv3x3_wmma(
    const float* __restrict__ X, int xbs,
    const float* __restrict__ Wt,                       // [Co][Cin][3][3]
    const float* __restrict__ scale, const float* __restrict__ bias,
    const float* __restrict__ resid, int rbs,
    const float* __restrict__ awmul,
    float* __restrict__ Y, int ybs,
    int Cin, int Co, int H, int W, int act)
{
    const int wave = threadIdx.x >> 5;
    const int lane = threadIdx.x & 31;
    const int l15  = lane & 15;
    const int hi   = lane >> 4;
    const int coBase = blockIdx.y * 64 + wave * 16;
    if (coBase >= Co) return;
    const int tilesW = (W + 15) >> 4;
    const int y   = blockIdx.x / tilesW;
    const int x0t = (blockIdx.x % tilesW) * 16;
    const int b = blockIdx.z;
    const int HWl = H * W;

    const float* __restrict__ Xb = X + (size_t)b * xbs;
    const int xcol = x0t + l15;
    const int kbase = hi * 2;
    const int mrow9 = (coBase + l15) * Cin * 9;
    const int bstep = 4 * HWl;

    v8f acc = {0,0,0,0,0,0,0,0};
#pragma unroll
    for (int t = 0; t < 9; ++t) {
        const int dy = t / 3 - 1, dx = t % 3 - 1;
        const int yy = y + dy;
        const int xx = xcol + dx;
        const float msk = (yy >= 0 && yy < H && xx >= 0 && xx < W) ? 1.0f : 0.0f;
        const int yyc = yy < 0 ? 0 : (yy >= H ? H - 1 : yy);
        const int xxc = xx < 0 ? 0 : (xx >= W ? W - 1 : xx);
        int aoff = mrow9 + t + kbase * 9;           // marches +36
        int boff = yyc * W + xxc + kbase * HWl;     // marches +4*HW
        for (int k0 = 0; k0 < Cin; k0 += 4) {
            v2f a, bf;
            a.x  = Wt[aoff];
            a.y  = Wt[aoff + 9];
            bf.x = Xb[boff] * msk;
            bf.y = Xb[boff + HWl] * msk;
            acc = __builtin_amdgcn_wmma_f32_16x16x4_f32(false, a, false, bf, (short)0, acc, false, false);
            aoff += 36;
            boff += bstep;
        }
    }
    if (xcol >= W) return;
    const int pix = y * W + xcol;
    float* __restrict__ Yb = Y + (size_t)b * ybs;
    const float am = awmul ? awmul[b * HWl + pix] : 1.0f;
#pragma unroll
    for (int r = 0; r < 8; ++r) {
        const int row = coBase + hi * 8 + r;
        float v = acc[r] * (scale ? scale[row] * BN_RS : 1.0f) + (bias ? bias[row] : 0.0f);
        if (act) v = silu_f(v);
        v *= am;
        if (resid) v += resid[(size_t)b * rbs + row * HWl + pix];
        Yb[row * HWl + pix] = v;
    }
}

// ===========================================================================
// aw[b,hw] = sigmoid(max_n(scores[b,n,hw]) / sqrt(128) + bias0)
// ===========================================================================
__global__ void maxsig_kernel(const float* __restrict__ scor, const float* __restrict__ bias0,
                              float* __restrict__ aw, int HW)
{
    int i = blockIdx.x * blockDim.x + threadIdx.x;
    int b = blockIdx.y;
    if (i >= HW) return;
    const float* s = scor + (size_t)b * 80 * HW + i;
    float m = s[0];
    for (int n = 1; n < 80; ++n) m = fmaxf(m, s[(size_t)n * HW]);
    float z = m * 0.08838834764831845f + bias0[0];
    aw[(size_t)b * HW + i] = 1.0f / (1.0f + __expf(-z));
}

// concat [p1 | p2 | m | attn] along channels (p1/p2 are x's own channels)
__global__ void concat4_kernel(const float* __restrict__ x, int C,
                               const float* __restrict__ m, const float* __restrict__ attn,
                               float* __restrict__ outc, int ch, int HW, long total)
{
    long tid = (long)blockIdx.x * blockDim.x + threadIdx.x;
    if (tid >= total) return;
    long per = (long)4 * ch * HW;
    long b = tid / per;
    long rem = tid - b * per;
    long c4 = rem / HW;
    long hw = rem - c4 * HW;
    float v;
    if (c4 < 2 * ch)      v = x[(b * C + c4) * HW + hw];
    else if (c4 < 3 * ch) v = m[(b * ch + (c4 - 2 * ch)) * HW + hw];
    else                  v = attn[(b * ch + (c4 - 3 * ch)) * HW + hw];
    outc[tid] = v;
}

// iterative top-k (ns <= 32) per batch; destroys aw
__global__ void topk_kernel(float* __restrict__ aw, int* __restrict__ idx, int HW, int ns)
{
    int b = blockIdx.x;
    float* a = aw + (size_t)b * HW;
    __shared__ float bv[256];
    __shared__ int   bi[256];
    for (int it = 0; it < ns; ++it) {
        float best = -1e30f; int besti = 0x7fffffff;
        for (int i = threadIdx.x; i < HW; i += 256) {
            float v = a[i];
            if (v > best) { best = v; besti = i; }
        }
        bv[threadIdx.x] = best; bi[threadIdx.x] = besti;
        __syncthreads();
        for (int s = 128; s > 0; s >>= 1) {
            if (threadIdx.x < s) {
                if (bv[threadIdx.x + s] > bv[threadIdx.x] ||
                    (bv[threadIdx.x + s] == bv[threadIdx.x] && bi[threadIdx.x + s] < bi[threadIdx.x])) {
                    bv[threadIdx.x] = bv[threadIdx.x + s];
                    bi[threadIdx.x] = bi[threadIdx.x + s];
                }
            }
            __syncthreads();
        }
        if (threadIdx.x == 0) { idx[b * ns + it] = bi[0]; a[bi[0]] = -1e30f; }
        __syncthreads();
    }
}

// gather top-k features + L2-normalized copy
__global__ void gcn_gather(const float* __restrict__ fused, long fbs, const int* __restrict__ idx,
                           float* __restrict__ feats, float* __restrict__ nf, int ns, int C, int HW)
{
    int i = blockIdx.x;          // 0..B*ns-1
    int b = i / ns;
    int pix = idx[i];
    __shared__ float red[256];
    float ss = 0.0f;
    for (int c = threadIdx.x; c < C; c += 256) {
        float v = fused[(size_t)b * fbs + (size_t)c * HW + pix];
        feats[(size_t)i * C + c] = v;
        ss += v * v;
    }
    red[threadIdx.x] = ss; __syncthreads();
    for (int s = 128; s > 0; s >>= 1) { if (threadIdx.x < s) red[threadIdx.x] += red[threadIdx.x + s]; __syncthreads(); }
    __shared__ float ninv;
    if (threadIdx.x == 0) ninv = 1.0f / fmaxf(sqrtf(red[0]), 1e-12f);
    __syncthreads();
    float nv = ninv;
    for (int c = threadIdx.x; c < C; c += 256) nf[(size_t)i * C + c] = feats[(size_t)i * C + c] * nv;
}

// thresholded cosine adjacency + row sums (Bns <= 64, blockDim = 64)
__global__ void gcn_adj(const float* __restrict__ nf, float* __restrict__ A,
                        float* __restrict__ dsum, int Bns, int C)
{
    int i = blockIdx.x;
    int j = threadIdx.x;
    __shared__ float red[64];
    float a = 0.0f;
    if (j < Bns) {
        float d = 0.0f;
        for (int c = 0; c < C; ++c) d += nf[(size_t)i * C + c] * nf[(size_t)j * C + c];
        a = (d > 0.5f) ? 1.0f : 0.0f;
        A[i * Bns + j] = a;
    }
    red[j] = a; __syncthreads();
    for (int s = 32; s > 0; s >>= 1) { if (j < s) red[j] += red[j + s]; __syncthreads(); }
    if (j == 0) dsum[i] = red[0];
}

__global__ void gcn_an(float* __restrict__ A, const float* __restrict__ dsum, int Bns)
{
    int t = blockIdx.x * blockDim.x + threadIdx.x;
    if (t >= Bns * Bns) return;
    int i = t / Bns, j = t % Bns;
    A[t] *= rsqrtf(dsum[i]) * rsqrtf(dsum[j]);
}

// Y[m,n] = (bias[n]) + sum_k A[m,k]*B[k,n] ; optional relu
__global__ void small_mm(const float* __restrict__ A, const float* __restrict__ B,
                         const float* __restrict__ bias, float* __restrict__ Yv,
                         int M, int K, int N, int relu)
{
    int id = blockIdx.x * blockDim.x + threadIdx.x;
    if (id >= M * N) return;
    int m = id / N, n = id % N;
    float s = bias ? bias[n] : 0.0f;
    for (int k = 0; k < K; ++k) s += A[(size_t)m * K + k] * B[(size_t)k * N + n];
    if (relu) s = fmaxf(s, 0.0f);
    Yv[id] = s;
}

__global__ void gcn_scatter(float* __restrict__ fused, long fbs, const int* __restrict__ idx,
                            const float* __restrict__ U, int ns, int C, int HW, int total)
{
    int t = blockIdx.x * blockDim.x + threadIdx.x;
    if (t >= total) return;
    int i = t / C, c = t % C;
    int b = i / ns;
    fused[(size_t)b * fbs + (size_t)c * HW + idx[i]] = U[(size_t)i * C + c];
}

// overlapping 3x3 adaptive max pool -> xc[b, level*9+ij, c]
__global__ void amax_pool3(const float* __restrict__ yp, float* __restrict__ xc,
                           int H, int W, int level)
{
    int t = blockIdx.x * blockDim.x + threadIdx.x;
    if (t >= 2 * 128 * 9) return;
    int b = t / (128 * 9);
    int r = t % (128 * 9);
    int c = r / 9;
    int ij = r % 9;
    int i = ij / 3, j = ij % 3;
    int hs = i * H / 3, he = ((i + 1) * H + 2) / 3;
    int wss = j * W / 3, we = ((j + 1) * W + 2) / 3;
    const float* p = yp + ((size_t)b * 128 + c) * H * W;
    float mx = -3.4e38f;
    for (int yy = hs; yy < he; ++yy)
        for (int xx = wss; xx < we; ++xx)
            mx = fmaxf(mx, p[(size_t)yy * W + xx]);
    xc[((size_t)b * 27 + level * 9 + ij) * 128 + c] = mx;
}

// LN(512) + 512->128 projection (q path). blockDim = 128.
__global__ __launch_bounds__(128) void ln_proj_q(const float* __restrict__ text,
    const float* __restrict__ g, const float* __restrict__ beta,
    const float* __restrict__ Wq, const float* __restrict__ bq, float* __restrict__ Q)
{
    int row = blockIdx.x;
    int t = threadIdx.x;
    const float* xr = text + (size_t)row * 512;
    __shared__ float sh[512];
    __shared__ float red[128];
    float s = 0.0f;
    for (int c = t; c < 512; c += 128) s += xr[c];
    red[t] = s; __syncthreads();
    for (int st = 64; st > 0; st >>= 1) { if (t < st) red[t] += red[t + st]; __syncthreads(); }
    float mu = red[0] / 512.0f; __syncthreads();
    float v = 0.0f;
    for (int c = t; c < 512; c += 128) { float d = xr[c] - mu; v += d * d; }
    red[t] = v; __syncthreads();
    for (int st = 64; st > 0; st >>= 1) { if (t < st) red[t] += red[t + st]; __syncthreads(); }
    float rs = rsqrtf(red[0] / 512.0f + LN_EPS); __syncthreads();
    for (int c = t; c < 512; c += 128) sh[c] = (xr[c] - mu) * rs * g[c] + beta[c];
    __syncthreads();
    float acc = bq[t];
    for (int c = 0; c < 512; ++c) acc += sh[c] * Wq[(size_t)t * 512 + c];
    Q[(size_t)row * 128 + t] = acc;
}

// LN(128) (shared mu/var, two affines) + two 128->128 projections (k,v). blockDim = 128.
__global__ __launch_bounds__(128) void ln_proj_kv(const float* __restrict__ xc,
    const float* __restrict__ kg, const float* __restrict__ kbln,
    const float* __restrict__ kw, const float* __restrict__ kb,
    const float* __restrict__ vg, const float* __restrict__ vbln,
    const float* __restrict__ vw, const float* __restrict__ vb,
    float* __restrict__ K, float* __restrict__ V)
{
    int row = blockIdx.x;
    int t = threadIdx.x;
    const float* xr = xc + (size_t)row * 128;
    __shared__ float nb[128];
    __shared__ float red[128];
    float val = xr[t];
    red[t] = val; __syncthreads();
    for (int s = 64; s > 0; s >>= 1) { if (t < s) red[t] += red[t + s]; __syncthreads(); }
    float mu = red[0] / 128.0f; __syncthreads();
    float d = val - mu;
    red[t] = d * d; __syncthreads();
    for (int s = 64; s > 0; s >>= 1) { if (t < s) red[t] += red[t + s]; __syncthreads(); }
    float rs = rsqrtf(red[0] / 128.0f + LN_EPS); __syncthreads();
    nb[t] = d * rs;
    __syncthreads();
    float ak = kb[t], av = vb[t];
    for (int c = 0; c < 128; ++c) {
        float n = nb[c];
        ak += (n * kg[c] + kbln[c]) * kw[(size_t)t * 128 + c];
        av += (n * vg[c] + vbln[c]) * vw[(size_t)t * 128 + c];
    }
    K[(size_t)row * 128 + t] = ak;
    V[(size_t)row * 128 + t] = av;
}

// per (b, query token, head): 27-key softmax attention, 16-dim output
__global__ void ipa_attn_kernel(const float* __restrict__ q, const float* __restrict__ K,
                                const float* __restrict__ V, float* __restrict__ o)
{
    int t = blockIdx.x * blockDim.x + threadIdx.x;
    if (t >= 2 * 80 * 8) return;
    int h = t % 8;
    int n = (t / 8) % 80;
    int b = t / (8 * 80);
    const float* qv = q + ((size_t)(b * 80 + n) * 128) + h * 16;
    float sc[27];
    float mx = -3.4e38f;
    for (int kt = 0; kt < 27; ++kt) {
        const float* kv = K + ((size_t)(b * 27 + kt) * 128) + h * 16;
        float dd = 0.0f;
        for (int c = 0; c < 16; ++c) dd += qv[c] * kv[c];
        dd *= 0.25f;                       // 1/sqrt(16)
        sc[kt] = dd;
        mx = fmaxf(mx, dd);
    }
    float se = 0.0f;
    for (int kt = 0; kt < 27; ++kt) { sc[kt] = __expf(sc[kt] - mx); se += sc[kt]; }
    float inv = 1.0f / se;
    for (int c = 0; c < 16; ++c) {
        float acc = 0.0f;
        for (int kt = 0; kt < 27; ++kt) acc += sc[kt] * V[((size_t)(b * 27 + kt) * 128) + h * 16 + c];
        o[((size_t)(b * 80 + n) * 128) + h * 16 + c] = acc * inv;
    }
}

// Y[r,co] = X[r,:]·W[co,:] + bias[co] (+ resid[r,co])
__global__ void linear_rowmajor(const float* __restrict__ X, const float* __restrict__ Wt,
                                const float* __restrict__ bias, const float* __restrict__ resid,
                                float* __restrict__ Y, int rows, int Ci, int Co)
{
    int t = blockIdx.x * blockDim.x + threadIdx.x;
    if (t >= rows * Co) return;
    int r = t / Co, co = t % Co;
    float a = bias ? bias[co] : 0.0f;
    for (int c = 0; c < Ci; ++c) a += X[(size_t)r * Ci + c] * Wt[(size_t)co * Ci + c];
    if (resid) a += resid[t];
    Y[t] = a;
}

// ===========================================================================
// Host side
// ===========================================================================
static const long OFF_M      = 0L;         // 1,638,400
static const long OFF_HID    = 1638400L;   //   819,200
static const long OFF_EMB    = 2457600L;   //   409,600
static const long OFF_ATTN   = 2867200L;   // 1,638,400
static const long OFF_CONCAT = 4505600L;   // 6,553,600 (reused for scores / yproj)
static const long OFF_AW     = 11059200L;  //    12,800
static const long OFF_T      = 11072000L;  //    20,480
static const long OFF_FEATS  = 11092480L;  //    16,384
static const long OFF_NF     = 11108864L;  //    16,384
static const long OFF_A      = 11125248L;  //     4,096
static const long OFF_DSUM   = 11129344L;  //        64
static const long OFF_T1     = 11129408L;  //     4,096
static const long OFF_H1     = 11133504L;  //     4,096
static const long OFF_T2     = 11137600L;  //    16,384
static const long OFF_U      = 11153984L;  //    16,384
static const long OFF_XC     = 11170368L;  //     6,912
static const long OFF_Q      = 11177280L;  //    20,480
static const long OFF_K      = 11197760L;  //     6,912
static const long OFF_V      = 11204672L;  //     6,912
static const long OFF_O      = 11211584L;  //    20,480
static const long OFF_IDX    = 11232064L;  //        64 ints

struct FU {
    const float *bias0, *img_b, *img_g, *img_w, *pc_b, *pc_g, *pc_w;
    const float *tl_b, *tl_w, *fus_b, *fus_g, *fus_w;
    const float *g1b, *g1w, *g2b, *g2w;
    const float *cv1_b, *cv1_g, *cv1_w, *cv2_b, *cv2_g, *cv2_w;
    bool has_img;
};

struct IPA {
    const float *k_b, *k_ln_b, *k_ln_g, *k_w;
    const float *proj_b0, *proj_b1, *proj_b2;
    const float *proj_out_b, *proj_out_w;
    const float *proj_w0, *proj_w1, *proj_w2;
    const float *q_b, *q_ln_b, *q_ln_g, *q_w;
    const float *v_b, *v_ln_b, *v_ln_g, *v_w;
};

static void run_fu(hipStream_t stream, const float* x, const float* textlike,
                   const FU& p, int C, int H, int W, float* out, float* ws)
{
    const int B = 2, ch = C / 2, hid = ch / 2, HW = H * W;
    float* mbuf = ws + OFF_M;
    float* hidb = ws + OFF_HID;
    float* embb = ws + OFF_EMB;
    float* attn = ws + OFF_ATTN;
    float* conc = ws + OFF_CONCAT;
    float* scor = ws + OFF_CONCAT;   // reused (time-disjoint with concat)
    float* aw   = ws + OFF_AW;
    float* tbuf = ws + OFF_T;
    int*   idxb = (int*)(ws + OFF_IDX);
    dim3 blk(128);

    // cv1: 1x1, p2 -> hid, SiLU
    conv1x1_wmma<<<dim3((HW + 63) / 64, (hid + 63) / 64, B), blk, 0, stream>>>(
        x + (size_t)ch * HW, C * HW, HW, 1,
        p.cv1_w, 0, p.cv1_g, p.cv1_b,
        hidb, hid * HW, HW, 1, ch, hid, HW, 1);

    // cv2: 3x3, hid -> ch, SiLU, + p2 residual  => m
    conv3x3_wmma<<<dim3(H * ((W + 15) / 16), (ch + 63) / 64, B), blk, 0, stream>>>(
        hidb, hid * HW, p.cv2_w, p.cv2_g, p.cv2_b,
        x + (size_t)ch * HW, C * HW, nullptr,
        mbuf, ch * HW, hid, ch, H, W, 1);

    // t = textlike @ tl_w^T + tl_b   (stored token-major (B,80,128))
    conv1x1_wmma<<<dim3((80 + 63) / 64, 2, B), blk, 0, stream>>>(
        textlike, 80 * 512, 1, 512, p.tl_w, 0, nullptr, p.tl_b,
        tbuf, 80 * 128, 1, 128, 512, 128, 80, 0);

    // emb
    const float* emb = mbuf; int embbs = ch * HW;
    if (p.has_img) {
        conv1x1_wmma<<<dim3((HW + 63) / 64, 2, B), blk, 0, stream>>>(
            mbuf, ch * HW, HW, 1, p.img_w, 0, p.img_g, p.img_b,
            embb, 128 * HW, HW, 1, ch, 128, HW, 0);
        emb = embb; embbs = 128 * HW;
    }

    // scores[b,n,hw] = sum_c emb[b,c,hw]*t[b,n,c]  (per-batch weight = t)
    conv1x1_wmma<<<dim3((HW + 63) / 64, 2, B), blk, 0, stream>>>(
        emb, embbs, HW, 1, tbuf, 80 * 128, nullptr, nullptr,
        scor, 80 * HW, HW, 1, 128, 80, HW, 0);

    maxsig_kernel<<<dim3((HW + 255) / 256, B), 256, 0, stream>>>(scor, p.bias0, aw, HW);

    // attn_feat = bn(proj_conv(m)) * aw
    conv3x3_wmma<<<dim3(H * ((W + 15) / 16), (ch + 63) / 64, B), blk, 0, stream>>>(
        mbuf, ch * HW, p.pc_w, p.pc_g, p.pc_b,
        nullptr, 0, aw,
        attn, ch * HW, ch, ch, H, W, 0);

    // concat + fusion -> out (d_out region)
    long tot = (long)B * 4 * ch * HW;
    concat4_kernel<<<(unsigned)((tot + 255) / 256), 256, 0, stream>>>(x, C, mbuf, attn, conc, ch, HW, tot);
    conv1x1_wmma<<<dim3((HW + 63) / 64, (C + 63) / 64, B), blk, 0, stream>>>(
        conc, 4 * ch * HW, HW, 1, p.fus_w, 0, p.fus_g, p.fus_b,
        out, C * HW, HW, 1, 4 * ch, C, HW, 0);

    // top-k + GCN refinement (scatter back into out)
    int ns = (int)(HW * 0.005);
    int Bns = B * ns;
    topk_kernel<<<B, 256, 0, stream>>>(aw, idxb, HW, ns);
    gcn_gather<<<Bns, 256, 0, stream>>>(out, (long)C * HW, idxb, ws + OFF_FEATS, ws + OFF_NF, ns, C, HW);
    gcn_adj<<<Bns, 64, 0, stream>>>(ws + OFF_NF, ws + OFF_A, ws + OFF_DSUM, Bns, C);
    gcn_an<<<(Bns * Bns + 63) / 64, 64, 0, stream>>>(ws + OFF_A, ws + OFF_DSUM, Bns);
    small_mm<<<(Bns * 64 + 255) / 256, 256, 0, stream>>>(ws + OFF_FEATS, p.g1w, nullptr, ws + OFF_T1, Bns, C, 64, 0);
    small_mm<<<(Bns * 64 + 255) / 256, 256, 0, stream>>>(ws + OFF_A, ws + OFF_T1, p.g1b, ws + OFF_H1, Bns, Bns, 64, 1);
    small_mm<<<(Bns * C + 255) / 256, 256, 0, stream>>>(ws + OFF_H1, p.g2w, nullptr, ws + OFF_T2, Bns, 64, C, 0);
    small_mm<<<(Bns * C + 255) / 256, 256, 0, stream>>>(ws + OFF_A, ws + OFF_T2, p.g2b, ws + OFF_U, Bns, Bns, C, 0);
    gcn_scatter<<<(Bns * C + 255) / 256, 256, 0, stream>>>(out, (long)C * HW, idxb, ws + OFF_U, ns, C, HW, Bns * C);
}

static void run_ipa(hipStream_t stream, const float* x0, const float* x1, const float* x2,
                    const float* textf, const IPA& p, float* tout, float* ws)
{
    const int B = 2;
    float* yp0 = ws + OFF_CONCAT;
    float* yp1 = yp0 + 2L * 128 * 6400;
    float* yp2 = yp1 + 2L * 128 * 1600;
    float* xc  = ws + OFF_XC;
    dim3 blk(128);

    conv1x1_wmma<<<dim3((6400 + 63) / 64, 2, B), blk, 0, stream>>>(
        x0, 256 * 6400, 6400, 1, p.proj_w0, 0, nullptr, p.proj_b0,
        yp0, 128 * 6400, 6400, 1, 256, 128, 6400, 0);
    conv1x1_wmma<<<dim3((1600 + 63) / 64, 2, B), blk, 0, stream>>>(
        x1, 512 * 1600, 1600, 1, p.proj_w1, 0, nullptr, p.proj_b1,
        yp1, 128 * 1600, 1600, 1, 512, 128, 1600, 0);
    conv1x1_wmma<<<dim3((400 + 63) / 64, 2, B), blk, 0, stream>>>(
        x2, 1024 * 400, 400, 1, p.proj_w2, 0, nullptr, p.proj_b2,
        yp2, 128 * 400, 400, 1, 1024, 128, 400, 0);

    amax_pool3<<<(2 * 128 * 9 + 255) / 256, 256, 0, stream>>>(yp0, xc, 80, 80, 0);
    amax_pool3<<<(2 * 128 * 9 + 255) / 256, 256, 0, stream>>>(yp1, xc, 40, 40, 1);
    amax_pool3<<<(2 * 128 * 9 + 255) / 256, 256, 0, stream>>>(yp2, xc, 20, 20, 2);

    ln_proj_q<<<160, 128, 0, stream>>>(textf, p.q_ln_g, p.q_ln_b, p.q_w, p.q_b, ws + OFF_Q);
    ln_proj_kv<<<54, 128, 0, stream>>>(xc, p.k_ln_g, p.k_ln_b, p.k_w, p.k_b,
                                       p.v_ln_g, p.v_ln_b, p.v_w, p.v_b, ws + OFF_K, ws + OFF_V);
    ipa_attn_kernel<<<(1280 + 127) / 128, 128, 0, stream>>>(ws + OFF_Q, ws + OFF_K, ws + OFF_V, ws + OFF_O);
    linear_rowmajor<<<(160 * 512 + 255) / 256, 256, 0, stream>>>(
        ws + OFF_O, p.proj_out_w, p.proj_out_b, textf, tout, 160, 128, 512);
}

extern "C" void kernel_launch(void* const* d_in, const int* in_sizes, int n_in,
                              void* d_out, int out_size, void* d_ws, size_t ws_size,
                              hipStream_t stream)
{
    const float* P3 = (const float*)d_in[0];
    const float* P4 = (const float*)d_in[1];
    const float* P5 = (const float*)d_in[2];
    const float* Q3 = (const float*)d_in[3];
    const float* Q4 = (const float*)d_in[4];
    const float* Q5 = (const float*)d_in[5];

    // text_feat is either at index 6 (insertion order) or last (sorted pytree order)
    const float* textf;
    int pi;
    if (in_sizes[6] == 2 * 80 * 512) { textf = (const float*)d_in[6]; pi = 7; }
    else                             { textf = (const float*)d_in[n_in - 1]; pi = 6; }

    auto nx = [&]() { return (const float*)d_in[pi++]; };

    // params in sorted-pytree leaf order: ipa, l3, l4, l5
    IPA ipa;
    ipa.k_b = nx(); ipa.k_ln_b = nx(); ipa.k_ln_g = nx(); ipa.k_w = nx();
    ipa.proj_b0 = nx(); ipa.proj_b1 = nx(); ipa.proj_b2 = nx();
    ipa.proj_out_b = nx(); ipa.proj_out_w = nx();
    ipa.proj_w0 = nx(); ipa.proj_w1 = nx(); ipa.proj_w2 = nx();
    ipa.q_b = nx(); ipa.q_ln_b = nx(); ipa.q_ln_g = nx(); ipa.q_w = nx();
    ipa.v_b = nx(); ipa.v_ln_b = nx(); ipa.v_ln_g = nx(); ipa.v_w = nx();

    auto readFU = [&](bool himg) {
        FU f; f.has_img = himg;
        f.bias0 = nx();
        if (himg) { f.img_b = nx(); f.img_g = nx(); f.img_w = nx(); }
        else      { f.img_b = f.img_g = f.img_w = nullptr; }
        f.pc_b = nx(); f.pc_g = nx(); f.pc_w = nx();
        f.tl_b = nx(); f.tl_w = nx();
        f.fus_b = nx(); f.fus_g = nx(); f.fus_w = nx();
        f.g1b = nx(); f.g1w = nx(); f.g2b = nx(); f.g2w = nx();
        f.cv1_b = nx(); f.cv1_g = nx(); f.cv1_w = nx();
        f.cv2_b = nx(); f.cv2_g = nx(); f.cv2_w = nx();
        return f;
    };
    FU l3 = readFU(false);
    FU l4 = readFU(true);
    FU l5 = readFU(true);

    float* ws  = (float*)d_ws;
    float* out = (float*)d_out;
    float* P3u = out;
    float* P4u = P3u + 3276800L;
    float* P5u = P4u + 1638400L;
    float* Q3u = P5u + 819200L;
    float* Q4u = Q3u + 3276800L;
    float* Q5u = Q4u + 1638400L;
    float* Tout = Q5u + 819200L;

    run_fu(stream, P3, textf, l3, 256, 80, 80, P3u, ws);
    run_fu(stream, Q3, textf, l3, 256, 80, 80, Q3u, ws);
    run_ipa(stream, P3u, P4, P5, textf, ipa, Tout, ws);
    run_fu(stream, P4, Tout, l4, 512, 40, 40, P4u, ws);
    run_fu(stream, P5, Tout, l5, 1024, 20, 20, P5u, ws);
    run_fu(stream, Q4, Tout, l4, 512, 40, 40, Q4u, ws);
    run_fu(stream, Q5, Tout, l5, 1024, 20, 20, Q5u, ws);
}